// QueryAndGroup_grouped_density_78374563217945
// MI455X (gfx1250) — compile-verified
//
#include <hip/hip_runtime.h>
#include <cstddef>
#include <cstdint>

// ---------------------------------------------------------------------------
// PointNet++ QueryAndGroup with density, CDNA5 (gfx1250) implementation.
//   d2(q, x) = |q|^2 + |x|^2 - 2 q.x   computed 16x16 per V_WMMA_F32_16X16X4_F32:
//     A row m  = [qx, qy, qz, |q|^2]        (16x4)
//     B col n  = [-2x, -2y, -2z, 1]         (4x16)
//     C        = |x|^2 broadcast per column
//   => D = A*B + C is the exact pairwise squared-distance tile.
// Candidate xyz points are streamed through LDS with a per-wave double-buffered
// GLOBAL_LOAD_ASYNC_TO_LDS_B128 pipeline (ASYNCcnt-tracked), so the WMMA loop
// reads only LDS. Ball-query ordered selection via per-row ballots of d2 < r^2.
// ---------------------------------------------------------------------------

typedef __attribute__((ext_vector_type(2))) float v2f;
typedef __attribute__((ext_vector_type(8))) float v8f;

#define NSAMPLE   32
#define RADIUS2   0.04f      // 0.2^2
#define DENS_A    4.0f       // 1/(2.5*0.1)
#define DENS_K    50.0f      // 1/(2*0.1^2)

#define CHUNK     256                 // candidate points per LDS chunk
#define CBYTES    (CHUNK * 12)        // 3072 bytes per chunk buffer
#define WPB       8                   // waves per block (256 threads)

// One wave handles 16 query points; streams candidates via async LDS chunks.
// DENSITY=true : query==xyz, accumulate inverse-density (writes invDens[B*M]).
// DENSITY=false: record first-32-in-order indices (writes idxOut[B*M*32]).
template <bool DENSITY>
__global__ __launch_bounds__(256) void bq_wmma_kernel(
    const float* __restrict__ xyz,   // (B, N, 3)
    const float* __restrict__ qpts,  // (B, M, 3)
    int N, int M,
    float* __restrict__ invDens,     // (B, M) if DENSITY
    int* __restrict__ idxOut)        // (B, M, 32) if !DENSITY
{
    __shared__ unsigned char ldsChunks[WPB * 2 * CBYTES];   // 48 KB / block

    const int flatWave      = (blockIdx.x * blockDim.x + threadIdx.x) >> 5;
    const int lane          = threadIdx.x & 31;
    const int wv            = (threadIdx.x >> 5) & (WPB - 1);
    const int tilesPerBatch = M >> 4;
    const int b             = flatWave / tilesPerBatch;
    const int qbase         = (flatWave % tilesPerBatch) << 4;
    const int m             = lane & 15;     // query row owned by this lane
    const bool hi           = lane >= 16;

    // Load the 16 query points (duplicated across half-waves).
    const float* qp = qpts + ((size_t)b * M + (size_t)(qbase + m)) * 3;
    const float qx = qp[0], qy = qp[1], qz = qp[2];
    const float qq = qx * qx + qy * qy + qz * qz;

    // A matrix (16x4 f32): lanes 0-15 hold K=0,1 ; lanes 16-31 hold K=2,3.
    v2f Amat;
    Amat.x = hi ? qz : qx;
    Amat.y = hi ? qq : qy;

    const float* xb = xyz + (size_t)b * N * 3;

    // LDS byte offset of this wave's two chunk buffers (low 32 bits of the
    // flat shared-variable address are the hardware LDS offset).
    unsigned char* wbuf = &ldsChunks[(size_t)wv * 2 * CBYTES];
    const unsigned ldsBase = (unsigned)(uintptr_t)wbuf;

    // Async-copy one 3 KB chunk of candidate coords into one LDS buffer:
    // 6 x (32 lanes x 16 B) byte-exact copies, tracked by ASYNCcnt.
    auto prefetch = [&](int chunkIdx, int bufSel) {
        const unsigned char* gsrc =
            (const unsigned char*)xb + (size_t)chunkIdx * CBYTES;
#pragma unroll
        for (int i = 0; i < CBYTES / 512; ++i) {
            unsigned la  = ldsBase + (unsigned)bufSel * CBYTES +
                           (unsigned)(i * 512 + lane * 16);
            uint64_t ga  = (uint64_t)(uintptr_t)gsrc +
                           (uint64_t)(i * 512 + lane * 16);
            asm volatile("global_load_async_to_lds_b128 %0, %1, off"
                         :: "v"(la), "v"(ga)
                         : "memory");
        }
    };

    int   cnt       = hi ? NSAMPLE : 0;   // idle upper lanes report "done"
    int   firstIdx  = 0;
    float firstDens = 0.0f;
    float densSum   = 0.0f;
    int*  myIdx     = DENSITY ? nullptr
                              : (idxOut + ((size_t)b * M + (size_t)(qbase + m)) * NSAMPLE);

    const int numChunks = N / CHUNK;
    prefetch(0, 0);

    bool done = false;
    for (int c = 0; c < numChunks && !done; ++c) {
        const int bufSel = c & 1;
        if (c + 1 < numChunks) {
            prefetch(c + 1, bufSel ^ 1);           // overlap next chunk's DMA
            asm volatile("s_wait_asynccnt 6" ::: "memory");  // current resident
        } else {
            asm volatile("s_wait_asynccnt 0" ::: "memory");
        }
        const float* cb  = (const float*)(wbuf + (size_t)bufSel * CBYTES);
        const int    n0c = c * CHUNK;

#pragma unroll 1
        for (int t = 0; t < CHUNK / 16; ++t) {
            // B matrix (4x16 f32) from LDS: lanes 0-15 K=0,1 ; lanes 16-31 K=2,3.
            const float* xp = cb + (size_t)(t * 16 + m) * 3;
            const float x0 = xp[0], x1 = xp[1], x2 = xp[2];
            const float xx = x0 * x0 + x1 * x1 + x2 * x2;
            v2f Bmat;
            Bmat.x = hi ? (-2.0f * x2) : (-2.0f * x0);
            Bmat.y = hi ? 1.0f         : (-2.0f * x1);
            v8f Cm = {xx, xx, xx, xx, xx, xx, xx, xx};   // |x_n|^2 per column

            v8f D = __builtin_amdgcn_wmma_f32_16x16x4_f32(
                /*neg_a=*/false, Amat, /*neg_b=*/false, Bmat,
                /*c_mod=*/(short)0, Cm, /*reuse_a=*/false, /*reuse_b=*/false);

            // Per-row in-radius masks. D element r: bits[15:0] -> row r,
            // bits[31:16] -> row r+8; bit index & 15 == column index.
            unsigned bal0 = __builtin_amdgcn_ballot_w32(D[0] < RADIUS2);
            unsigned bal1 = __builtin_amdgcn_ballot_w32(D[1] < RADIUS2);
            unsigned bal2 = __builtin_amdgcn_ballot_w32(D[2] < RADIUS2);
            unsigned bal3 = __builtin_amdgcn_ballot_w32(D[3] < RADIUS2);
            unsigned bal4 = __builtin_amdgcn_ballot_w32(D[4] < RADIUS2);
            unsigned bal5 = __builtin_amdgcn_ballot_w32(D[5] < RADIUS2);
            unsigned bal6 = __builtin_amdgcn_ballot_w32(D[6] < RADIUS2);
            unsigned bal7 = __builtin_amdgcn_ballot_w32(D[7] < RADIUS2);

            // Lane m picks mask for row m: sel = bal[m & 7], half = (m >> 3).
            unsigned ta = (lane & 1) ? bal1 : bal0;
            unsigned tb = (lane & 1) ? bal3 : bal2;
            unsigned tc = (lane & 1) ? bal5 : bal4;
            unsigned td = (lane & 1) ? bal7 : bal6;
            unsigned te = (lane & 2) ? tb : ta;
            unsigned tf = (lane & 2) ? td : tc;
            unsigned tg = (lane & 4) ? tf : te;
            unsigned mask = (tg >> ((lane & 8) << 1)) & 0xffffu;

            // Ordered extraction (index order == bit order); gathers hit LDS.
            while (mask != 0u && cnt < NSAMPLE) {
                const int col = __builtin_ctz(mask);
                mask &= mask - 1u;
                const int lj = t * 16 + col;      // index within chunk
                const int j  = n0c + lj;          // global candidate index
                if (DENSITY) {
                    const float* pj = cb + (size_t)lj * 3;
                    const float dx = pj[0] - qx;
                    const float dy = pj[1] - qy;
                    const float dz = pj[2] - qz;
                    const float d2 = dx * dx + dy * dy + dz * dz;
                    const float w  = DENS_A * __expf(-DENS_K * d2);
                    if (cnt == 0) firstDens = w;
                    densSum += w;
                } else {
                    myIdx[cnt] = j;
                }
                if (cnt == 0) firstIdx = j;
                ++cnt;
            }

            // Uniform early-exit once every owned query has 32 neighbors.
            const unsigned notDone =
                __builtin_amdgcn_ballot_w32(cnt < NSAMPLE) & 0xffffu;
            if (notDone == 0u) { done = true; break; }
        }
    }

    if (!hi) {
        if (DENSITY) {
            if (cnt == 0) {  // reference falls back to index 0
                const float dx = xb[0] - qx, dy = xb[1] - qy, dz = xb[2] - qz;
                firstDens = DENS_A * __expf(-DENS_K * (dx * dx + dy * dy + dz * dz));
            }
            const float total = densSum + (float)(NSAMPLE - cnt) * firstDens;
            // grouped_density = 1 / mean(dens) = NSAMPLE / sum
            invDens[(size_t)b * M + (size_t)(qbase + m)] = (float)NSAMPLE / total;
        } else {
            // Unfilled slots replicate first found index (0 if none found).
            for (int k = cnt; k < NSAMPLE; ++k) myIdx[k] = firstIdx;
        }
    }
}

// Coalesced gather: out[b, c, q, k] with channels [xyz(3) | invDens(1) | feat(64)].
__global__ __launch_bounds__(256) void group_gather_kernel(
    const float* __restrict__ xyz,       // (B, N, 3)
    const float* __restrict__ invDens,   // (B, N)
    const float* __restrict__ features,  // (B, Cf, N)
    const int*   __restrict__ idx,       // (B, Q, 32)
    float* __restrict__ out,             // (B, Cf+4, Q, 32)
    int N, int Q, int Cf)
{
    const int bq = blockIdx.x;           // one (b, q) per block
    const int b  = bq / Q;
    const int q  = bq % Q;
    const int k  = threadIdx.x & 31;
    const int c0 = threadIdx.x >> 5;     // 8 channel slices per pass
    const int Ct = Cf + 4;

    const int j = idx[((size_t)b * Q + q) * NSAMPLE + k];
    const float* xb = xyz      + (size_t)b * N * 3;
    const float* fb = features + (size_t)b * Cf * N;

    for (int c = c0; c < Ct; c += 8) {
        float v;
        if (c < 3)       v = xb[(size_t)j * 3 + c];
        else if (c == 3) v = invDens[(size_t)b * N + j];
        else             v = fb[(size_t)(c - 4) * N + j];
        out[(((size_t)b * Ct + c) * Q + q) * NSAMPLE + k] = v;
    }
}

extern "C" void kernel_launch(void* const* d_in, const int* in_sizes, int n_in,
                              void* d_out, int out_size, void* d_ws, size_t ws_size,
                              hipStream_t stream) {
    const float* xyz      = (const float*)d_in[0];  // (2, 8192, 3)
    const float* new_xyz  = (const float*)d_in[1];  // (2, 2048, 3)
    const float* features = (const float*)d_in[2];  // (2, 64, 8192)

    const int B  = 2;
    const int N  = in_sizes[0] / (B * 3);           // 8192
    const int Q  = in_sizes[1] / (B * 3);           // 2048
    const int Cf = in_sizes[2] / (B * N);           // 64

    // Workspace: invDensity (B*N floats) then neighbor indices (B*Q*32 ints).
    float* invDens = (float*)d_ws;
    int*   idx     = (int*)((char*)d_ws + (size_t)B * N * sizeof(float));

    // Pass 1: density self-query. waves = B*(N/16); 8 waves/block.
    {
        const int waves  = B * (N / 16);
        const int blocks = (waves * 32 + 255) / 256;
        bq_wmma_kernel<true><<<blocks, 256, 0, stream>>>(
            xyz, xyz, N, N, invDens, nullptr);
    }
    // Pass 2: ball query of new_xyz against xyz, record indices.
    {
        const int waves  = B * (Q / 16);
        const int blocks = (waves * 32 + 255) / 256;
        bq_wmma_kernel<false><<<blocks, 256, 0, stream>>>(
            xyz, new_xyz, N, Q, nullptr, idx);
    }
    // Pass 3: gather + concat into (B, Cf+4, Q, 32).
    group_gather_kernel<<<B * Q, 256, 0, stream>>>(
        xyz, invDens, features, idx, (float*)d_out, N, Q, Cf);
}